// GNNModel_62921270886996
// MI455X (gfx1250) — compile-verified
//
#include <hip/hip_runtime.h>

#define N_NODES 50000
#define N_EDGES 320000
#define DIM     256

typedef __attribute__((ext_vector_type(2))) float v2f;
typedef __attribute__((ext_vector_type(8))) float v8f;

#if defined(__has_builtin)
#if __has_builtin(__builtin_amdgcn_wmma_f32_16x16x4_f32)
#define HAS_WMMA_F32_16X16X4 1
#endif
#endif

// ---------------------------------------------------------------------------
// Kernel 1: deg[n] = 1.0 (self-loop weight)
// ---------------------------------------------------------------------------
__global__ __launch_bounds__(256) void k_deg_init(float* __restrict__ deg) {
    int n = blockIdx.x * blockDim.x + threadIdx.x;
    if (n < N_NODES) deg[n] = 1.0f;
}

// ---------------------------------------------------------------------------
// Kernel 2: deg[col[e]] += edge_attr[e]
// ---------------------------------------------------------------------------
__global__ __launch_bounds__(256) void k_deg_accum(const int* __restrict__ ei,
                                                   const float* __restrict__ ea,
                                                   float* __restrict__ deg) {
    int e = blockIdx.x * blockDim.x + threadIdx.x;
    if (e >= N_EDGES) return;
    int c = ei[N_EDGES + e];  // edge_index[1][e]
    __hip_atomic_fetch_add(&deg[c], ea[e], __ATOMIC_RELAXED, __HIP_MEMORY_SCOPE_AGENT);
}

// ---------------------------------------------------------------------------
// Kernel 3: deg[n] = deg[n] > 0 ? rsqrt(deg[n]) : 0   (in place -> dinv)
// ---------------------------------------------------------------------------
__global__ __launch_bounds__(256) void k_dinv(float* __restrict__ deg) {
    int n = blockIdx.x * blockDim.x + threadIdx.x;
    if (n >= N_NODES) return;
    float d = deg[n];
    deg[n] = (d > 0.0f) ? rsqrtf(d) : 0.0f;
}

// ---------------------------------------------------------------------------
// Kernel 4: h = x @ W^T  via V_WMMA_F32_16X16X4_F32 (fp32, matches reference)
//
// Per wave: one 16(M) x 64(N) tile, 4 x v8f accumulators, K-loop stepping 4.
// A fragment (32-bit A-matrix 16x4, ISA layout):
//   lane<16:  v0=A[M=lane,K=k+0], v1=A[M=lane,K=k+1]
//   lane>=16: v0=A[M=lane-16,K=k+2], v1=A[M=lane-16,K=k+3]
// B fragment (4x16): B[k][n] = W[n][k]; same (lane&15 , lane>>4) pattern.
// Both are a single 8-byte float2 load per fragment per lane.
// C/D layout: VGPR i -> M = i + (lane>=16 ? 8:0), N = lane&15.
// ---------------------------------------------------------------------------
#if HAS_WMMA_F32_16X16X4
__global__ __launch_bounds__(256) void k_gemm_wmma(const float* __restrict__ x,
                                                   const float* __restrict__ W,
                                                   float* __restrict__ h) {
    const int wave = blockIdx.x * (blockDim.x >> 5) + (threadIdx.x >> 5);
    const int lane = threadIdx.x & 31;
    const int n_mtiles = N_NODES / 16;   // 3125 (exact)
    const int n_ntiles = DIM / 64;       // 4
    if (wave >= n_mtiles * n_ntiles) return;

    const int m0 = (wave / n_ntiles) * 16;
    const int n0 = (wave % n_ntiles) * 64;
    const int lhalf = lane >> 4;   // 0 or 1
    const int l15   = lane & 15;

    const float* arow = x + (long long)(m0 + l15) * DIM;
    const float* b0p  = W + (long long)(n0 +  0 + l15) * DIM;
    const float* b1p  = W + (long long)(n0 + 16 + l15) * DIM;
    const float* b2p  = W + (long long)(n0 + 32 + l15) * DIM;
    const float* b3p  = W + (long long)(n0 + 48 + l15) * DIM;

    v8f acc0 = {}, acc1 = {}, acc2 = {}, acc3 = {};

    for (int k = 0; k < DIM; k += 4) {
        const int kk = k + lhalf * 2;
        v2f a  = *(const v2f*)(arow + kk);
        v2f b0 = *(const v2f*)(b0p + kk);
        v2f b1 = *(const v2f*)(b1p + kk);
        v2f b2 = *(const v2f*)(b2p + kk);
        v2f b3 = *(const v2f*)(b3p + kk);
        acc0 = __builtin_amdgcn_wmma_f32_16x16x4_f32(false, a, false, b0, (short)0, acc0, false, false);
        acc1 = __builtin_amdgcn_wmma_f32_16x16x4_f32(false, a, false, b1, (short)0, acc1, false, false);
        acc2 = __builtin_amdgcn_wmma_f32_16x16x4_f32(false, a, false, b2, (short)0, acc2, false, false);
        acc3 = __builtin_amdgcn_wmma_f32_16x16x4_f32(false, a, false, b3, (short)0, acc3, false, false);
    }

    const int srow = m0 + lhalf * 8;
    for (int i = 0; i < 8; ++i) {
        float* hr = h + (long long)(srow + i) * DIM + n0 + l15;
        hr[ 0] = acc0[i];
        hr[16] = acc1[i];
        hr[32] = acc2[i];
        hr[48] = acc3[i];
    }
}
#else
// Fallback (no f32 WMMA builtin): one thread per output element.
__global__ __launch_bounds__(256) void k_gemm_wmma(const float* __restrict__ x,
                                                   const float* __restrict__ W,
                                                   float* __restrict__ h) {
    long long t = (long long)blockIdx.x * blockDim.x + threadIdx.x;
    if (t >= (long long)N_NODES * DIM) return;
    int m = (int)(t / DIM), n = (int)(t % DIM);
    float acc = 0.0f;
    for (int k = 0; k < DIM; ++k) acc = fmaf(x[(long long)m * DIM + k], W[(long long)n * DIM + k], acc);
    h[t] = acc;
}
#endif

// ---------------------------------------------------------------------------
// Kernel 5: agg[n,d] = b[d] + h[n,d] * dinv[n]^2   (self-loop msg + bias)
// float4-vectorized; writes into d_out which serves as the accumulator.
// ---------------------------------------------------------------------------
__global__ __launch_bounds__(256) void k_agg_init(const float* __restrict__ h,
                                                  const float* __restrict__ dinv,
                                                  const float* __restrict__ bias,
                                                  float* __restrict__ agg) {
    long long t = (long long)blockIdx.x * blockDim.x + threadIdx.x;  // N*DIM/4 elems
    if (t >= (long long)N_NODES * (DIM / 4)) return;
    int n = (int)(t / (DIM / 4));
    int d = (int)(t % (DIM / 4)) * 4;
    float s = dinv[n];
    s = s * s;
    float4 hv = *(const float4*)(h + (long long)n * DIM + d);
    float4 bv = *(const float4*)(bias + d);
    float4 o;
    o.x = bv.x + hv.x * s;
    o.y = bv.y + hv.y * s;
    o.z = bv.z + hv.z * s;
    o.w = bv.w + hv.w * s;
    *(float4*)(agg + (long long)n * DIM + d) = o;
}

// ---------------------------------------------------------------------------
// Kernel 6: edge scatter. One wave32 per edge; edge index forced wave-uniform
// via readfirstlane so ei/ea/dinv lookups become scalar (SMEM) loads done
// once per wave instead of per-lane gathers. Each lane handles 8 floats
// (two float4 segments, 512B coalesced chunks per half-row).
// agg[col] += h[row] * (dinv[row] * ea * dinv[col])
// ---------------------------------------------------------------------------
__global__ __launch_bounds__(256) void k_scatter(const int* __restrict__ ei,
                                                 const float* __restrict__ ea,
                                                 const float* __restrict__ dinv,
                                                 const float* __restrict__ h,
                                                 float* __restrict__ agg) {
    const int wave_g = blockIdx.x * (blockDim.x >> 5) + (threadIdx.x >> 5);
    const int e = __builtin_amdgcn_readfirstlane(wave_g);   // wave-uniform edge id
    if (e >= N_EDGES) return;
    const int lane = threadIdx.x & 31;

    const int r = ei[e];                 // scalar load (uniform address)
    const int c = ei[N_EDGES + e];       // scalar load
    const float norm = dinv[r] * ea[e] * dinv[c];  // all uniform -> scalar ops

    const float* hrow = h + (long long)r * DIM + lane * 4;
    float*       dst  = agg + (long long)c * DIM + lane * 4;

    float4 h0 = *(const float4*)(hrow);
    float4 h1 = *(const float4*)(hrow + 128);

    __hip_atomic_fetch_add(dst + 0,   h0.x * norm, __ATOMIC_RELAXED, __HIP_MEMORY_SCOPE_AGENT);
    __hip_atomic_fetch_add(dst + 1,   h0.y * norm, __ATOMIC_RELAXED, __HIP_MEMORY_SCOPE_AGENT);
    __hip_atomic_fetch_add(dst + 2,   h0.z * norm, __ATOMIC_RELAXED, __HIP_MEMORY_SCOPE_AGENT);
    __hip_atomic_fetch_add(dst + 3,   h0.w * norm, __ATOMIC_RELAXED, __HIP_MEMORY_SCOPE_AGENT);
    __hip_atomic_fetch_add(dst + 128, h1.x * norm, __ATOMIC_RELAXED, __HIP_MEMORY_SCOPE_AGENT);
    __hip_atomic_fetch_add(dst + 129, h1.y * norm, __ATOMIC_RELAXED, __HIP_MEMORY_SCOPE_AGENT);
    __hip_atomic_fetch_add(dst + 130, h1.z * norm, __ATOMIC_RELAXED, __HIP_MEMORY_SCOPE_AGENT);
    __hip_atomic_fetch_add(dst + 131, h1.w * norm, __ATOMIC_RELAXED, __HIP_MEMORY_SCOPE_AGENT);
}

// ---------------------------------------------------------------------------
// Kernel 7: out = x + relu(agg)   (in place on d_out)
// ---------------------------------------------------------------------------
__global__ __launch_bounds__(256) void k_final(const float* __restrict__ x,
                                               float* __restrict__ out) {
    long long t = (long long)blockIdx.x * blockDim.x + threadIdx.x;
    if (t >= (long long)N_NODES * (DIM / 4)) return;
    float4 a = *(const float4*)(out + t * 4);
    float4 xv = *(const float4*)(x + t * 4);
    float4 o;
    o.x = xv.x + fmaxf(a.x, 0.0f);
    o.y = xv.y + fmaxf(a.y, 0.0f);
    o.z = xv.z + fmaxf(a.z, 0.0f);
    o.w = xv.w + fmaxf(a.w, 0.0f);
    *(float4*)(out + t * 4) = o;
}

// ---------------------------------------------------------------------------
extern "C" void kernel_launch(void* const* d_in, const int* in_sizes, int n_in,
                              void* d_out, int out_size, void* d_ws, size_t ws_size,
                              hipStream_t stream) {
    const float* x    = (const float*)d_in[0];
    const int*   ei   = (const int*)d_in[1];    // [2, E] flattened
    const float* ea   = (const float*)d_in[2];
    const float* W    = (const float*)d_in[3];
    const float* bias = (const float*)d_in[4];
    float* out = (float*)d_out;

    // Workspace: deg/dinv (N floats = 200000 B, 16B-aligned), then h (N*DIM floats)
    float* deg = (float*)d_ws;
    float* h   = deg + N_NODES;

    const int nodes_blk = (N_NODES + 255) / 256;
    const int edges_blk = (N_EDGES + 255) / 256;

    k_deg_init<<<nodes_blk, 256, 0, stream>>>(deg);
    k_deg_accum<<<edges_blk, 256, 0, stream>>>(ei, ea, deg);
    k_dinv<<<nodes_blk, 256, 0, stream>>>(deg);

#if HAS_WMMA_F32_16X16X4
    const int n_waves = (N_NODES / 16) * (DIM / 64);      // 12500 waves
    const int gemm_blk = (n_waves + 7) / 8;               // 8 waves / 256-thr block
#else
    const int gemm_blk = (int)(((long long)N_NODES * DIM + 255) / 256);
#endif
    k_gemm_wmma<<<gemm_blk, 256, 0, stream>>>(x, W, h);

    const long long vec_elems = (long long)N_NODES * (DIM / 4);  // 3.2M
    const int vec_blk = (int)((vec_elems + 255) / 256);
    k_agg_init<<<vec_blk, 256, 0, stream>>>(h, deg, bias, out);

    const int sc_waves_per_blk = 8;                               // 256 thr = 8 waves
    const int sc_blk = (N_EDGES + sc_waves_per_blk - 1) / sc_waves_per_blk;  // 40000
    k_scatter<<<sc_blk, 256, 0, stream>>>(ei, ea, deg, h, out);

    k_final<<<vec_blk, 256, 0, stream>>>(x, out);
}